// AbsolutePositionEncoding_29540785062638
// MI455X (gfx1250) — compile-verified
//
#include <hip/hip_runtime.h>
#include <hip/hip_bf16.h>

typedef unsigned int       u32;
typedef unsigned long long u64;
typedef __attribute__((ext_vector_type(4))) u32 v4u;
typedef __attribute__((ext_vector_type(4))) int v4i;
typedef __attribute__((ext_vector_type(8))) int v8i;

#define PE_B 8
#define PE_H 2048
#define PE_W 2048
#define ROWS_PER_WG 8
#define NTHREADS 256
#define QUADS_PER_ROW  (PE_W / 4)                    // 512
#define QUADS_PER_TILE (ROWS_PER_WG * QUADS_PER_ROW) // 4096 quads = 64 KB
#define ITERS (QUADS_PER_TILE / NTHREADS)            // 16

#if defined(__HIP_DEVICE_COMPILE__) && __has_builtin(__builtin_amdgcn_tensor_store_from_lds)
#define HAVE_TDM 1
#else
#define HAVE_TDM 0
#endif

#if HAVE_TDM
// Issue one TDM store: 2-D tile (tile_rows x 2048 f32) from LDS offset 0 to
// global address `gaddr_bytes`, row stride 2048 elements. D# layout per
// CDNA5 ISA.
__device__ __forceinline__ void tdm_store_tile(u32 lds_byte_off, u64 gaddr_bytes,
                                               int tile_rows) {
  v4u g0;
  g0.x = 1u;                                            // count=1, is_restore=0, gather off
  g0.y = lds_byte_off;                                  // lds_addr (bytes, wave-relative)
  g0.z = (u32)(gaddr_bytes & 0xFFFFFFFFull);            // global_addr[31:0]
  g0.w = (u32)((gaddr_bytes >> 32) & 0x1FFFFFFull)      // global_addr[56:32]
         | (2u << 30);                                  // type = 2 ("image")
  v8i g1;
  g1[0] = (int)(2u << 16);                              // wg_mask=0, data_size=2 (4 B)
  g1[1] = (int)((u32)PE_W << 16);                       // tensor_dim0[15:0] @ bits[31:16]
  g1[2] = (int)((u32)PE_H << 16);                       // dim0 hi=0 | tensor_dim1[15:0]
  g1[3] = (int)((u32)PE_W << 16);                       // dim1 hi=0 | tile_dim0 = 2048
  g1[4] = tile_rows;                                    // tile_dim1 = rows, tile_dim2 = 0
  g1[5] = PE_W;                                         // tensor_dim0_stride lo32 = 2048
  g1[6] = 0;                                            // stride hi | dim1_stride lo (unused, 2-D)
  g1[7] = 0;
  v4i g2 = {0, 0, 0, 0};                                // 2-D tile: groups 2/3 unused
  v4i g3 = {0, 0, 0, 0};
#if __clang_major__ >= 23
  v8i g4 = {0, 0, 0, 0, 0, 0, 0, 0};
  __builtin_amdgcn_tensor_store_from_lds(g0, g1, g2, g3, g4, 0);
#else
  __builtin_amdgcn_tensor_store_from_lds(g0, g1, g2, g3, 0);
#endif
}
#endif // HAVE_TDM

__global__ __launch_bounds__(NTHREADS)
void AbsolutePositionEncoding_kernel(float* __restrict__ out) {
  extern __shared__ float4 tile[];   // QUADS_PER_TILE float4 = 64 KB
  const int t    = (int)threadIdx.x;
  const int row0 = (int)blockIdx.x * ROWS_PER_WG;

  // w_k = 10000^(-2j/2048) = exp2(c * j),  c = -log2(10000)/1024
  const float c      = -13.287712379549449f / 1024.0f;
  const float inv2pi = 0.15915494309189535f;

#pragma unroll
  for (int k = 0; k < ITERS; ++k) {
    const int   lin = k * NTHREADS + t;         // quad index within tile
    const int   r   = lin >> 9;                 // row within tile (512 quads/row)
    const int   q   = lin & (QUADS_PER_ROW - 1);
    const float row = (float)(row0 + r);
    const float j0  = (float)(4 * q);

    const float w0 = __builtin_amdgcn_exp2f(c * (j0 + 0.0f));
    const float w1 = __builtin_amdgcn_exp2f(c * (j0 + 1.0f));
    const float w2 = __builtin_amdgcn_exp2f(c * (j0 + 2.0f));
    const float w3 = __builtin_amdgcn_exp2f(c * (j0 + 3.0f));

    // Explicit revolution-domain range reduction, then raw v_sin/v_cos
    // (hw computes sin(2*pi*x); angles reach 2047 rad so fract first).
    const float t0 = __builtin_amdgcn_fractf(row * w0 * inv2pi);
    const float t1 = __builtin_amdgcn_fractf(row * w1 * inv2pi);
    const float t2 = __builtin_amdgcn_fractf(row * w2 * inv2pi);
    const float t3 = __builtin_amdgcn_fractf(row * w3 * inv2pi);

    float4 v;
    v.x = __builtin_amdgcn_sinf(t0);   // even column -> sin
    v.y = __builtin_amdgcn_cosf(t1);   // odd  column -> cos
    v.z = __builtin_amdgcn_sinf(t2);
    v.w = __builtin_amdgcn_cosf(t3);

#if HAVE_TDM
    tile[lin] = v;                     // ds_store_b128, conflict-free layout
#else
    const u64 eoff = (u64)(row0 + r) * PE_W + (u64)(4 * q);
#pragma unroll
    for (int b = 0; b < PE_B; ++b)
      *(float4*)(out + (u64)b * PE_H * PE_W + eoff) = v;
#endif
  }

#if HAVE_TDM
  __syncthreads();                      // all ds writes visible before DMA
  if (t == 0) {                         // wave 0 issues the DMA ops (EXEC ignored by TDM)
#pragma unroll
    for (int b = 0; b < PE_B; ++b) {
      const u64 ga = (u64)(uintptr_t)out +
                     (((u64)b * PE_H + (u64)row0) * (u64)PE_W) * 4ull;
      tdm_store_tile(0u, ga, ROWS_PER_WG);
    }
#if __has_builtin(__builtin_amdgcn_s_wait_tensorcnt)
    __builtin_amdgcn_s_wait_tensorcnt(0);
#endif
  }
#endif
}

extern "C" void kernel_launch(void* const* d_in, const int* in_sizes, int n_in,
                              void* d_out, int out_size, void* d_ws, size_t ws_size,
                              hipStream_t stream) {
  (void)d_in; (void)in_sizes; (void)n_in; (void)d_ws; (void)ws_size; (void)out_size;
  float* out = (float*)d_out;
  dim3 grid(PE_H / ROWS_PER_WG);      // 256 workgroups, one 8-row stripe each
  dim3 block(NTHREADS);
  size_t shmem = (size_t)QUADS_PER_TILE * sizeof(float4);  // 64 KB dynamic LDS
  AbsolutePositionEncoding_kernel<<<grid, block, shmem, stream>>>(out);
}